// EMDLoss_48447231099486
// MI455X (gfx1250) — compile-verified
//
#include <hip/hip_runtime.h>

typedef __attribute__((ext_vector_type(2))) float v2f;
typedef __attribute__((ext_vector_type(8))) float v8f;

#define BB 8
#define DD 3
#define NN 2048
#define EPSV 0.001f
#define INV_EPS 1000.0f
#define LOG_MU (-7.6246189861593985f)   // -log(2048)
#define TILES (NN / 16)
#define WAVES_PB 4
#define THREADS_PB (WAVES_PB * 32)
#define ROWS_PB (WAVES_PB * 16)        // 64 rows per block
#define MAX_ITER 200

__device__ __forceinline__ float half16_max(float v) {
    v = fmaxf(v, __shfl_xor(v, 1, 32));
    v = fmaxf(v, __shfl_xor(v, 2, 32));
    v = fmaxf(v, __shfl_xor(v, 4, 32));
    v = fmaxf(v, __shfl_xor(v, 8, 32));
    return v;
}
__device__ __forceinline__ float half16_sum(float v) {
    v += __shfl_xor(v, 1, 32);
    v += __shfl_xor(v, 2, 32);
    v += __shfl_xor(v, 4, 32);
    v += __shfl_xor(v, 8, 32);
    return v;
}

__global__ void emd_init(float* p, int n) {
    int i = blockIdx.x * blockDim.x + threadIdx.x;
    if (i < n) p[i] = 0.0f;
}

// One half Sinkhorn step. Row side (rowPts, uRow) updated in place; col side read-only.
// M_ij = (u_i + v_j - C_ij)/eps.  WMMA computes (2/eps)*x.y + beta_j via the K=3 slot
// (A_K3 = 1, B_K3 = beta_j); alpha_i rides in the loop-invariant C accumulator.
__global__ __launch_bounds__(THREADS_PB)
void sinkhorn_half_pass(const float* __restrict__ rowPts,
                        const float* __restrict__ colPts,
                        float* __restrict__ uRow,
                        const float* __restrict__ vCol) {
    __shared__ float4 sB[NN];   // (y0, y1 | y2, beta_j) -> two float2 halves
    const int b = blockIdx.y;
    const float* R  = rowPts + (size_t)b * DD * NN;
    const float* Cp = colPts + (size_t)b * DD * NN;
    float* u       = uRow + (size_t)b * NN;
    const float* v = vCol + (size_t)b * NN;

    for (int j = threadIdx.x; j < NN; j += blockDim.x) {
        float y0 = Cp[j], y1 = Cp[NN + j], y2 = Cp[2 * NN + j];
        float yn = y0 * y0 + y1 * y1 + y2 * y2;
        sB[j] = make_float4(y0, y1, y2, (v[j] - yn) * INV_EPS);  // .w = beta_j
    }
    __syncthreads();

    const int lane = threadIdx.x & 31;
    const int wave = threadIdx.x >> 5;
    const int i0   = blockIdx.x * ROWS_PB + wave * 16;
    const bool hi  = lane >= 16;
    const int rl   = lane & 15;

    // B operand: lo lanes take (y0,y1) = K0,K1; hi lanes take (y2,beta) = K2,K3.
    const v2f* bview = (const v2f*)sB;
    const int  bidx0 = 2 * rl + (hi ? 1 : 0);

    // A-matrix 16x4 (MxK): lo lanes (s*x0, s*x1); hi lanes (s*x2, 1.0) with s = 2/eps.
    v2f A;
    v8f afv;                       // C init: alpha_i per C/D-layout register
    {
        const int i = i0 + rl;
        const float x0 = R[i], x1 = R[NN + i], x2 = R[2 * NN + i];
        if (!hi) { A.x = x0 * (2.0f * INV_EPS); A.y = x1 * (2.0f * INV_EPS); }
        else     { A.x = x2 * (2.0f * INV_EPS); A.y = 1.0f; }
        const float xn = x0 * x0 + x1 * x1 + x2 * x2;
        const float aval = (u[i] - xn) * INV_EPS;           // alpha_i
        #pragma unroll
        for (int r = 0; r < 8; ++r)
            afv[r] = __shfl(aval, hi ? (r + 8) : r, 32);    // row r+8*half
    }

    // ---- pass 1: per-lane running max (1 ds_load_b64 + 1 wmma + 8 max per tile) ----
    float rMax[8];
    #pragma unroll
    for (int r = 0; r < 8; ++r) rMax[r] = -3.0e38f;

    for (int t = 0; t < TILES; ++t) {
        const v2f Bm = bview[bidx0 + 32 * t];
        const v8f acc = __builtin_amdgcn_wmma_f32_16x16x4_f32(false, A, false, Bm,
                                                              (short)0, afv, false, false);
        #pragma unroll
        for (int r = 0; r < 8; ++r) rMax[r] = fmaxf(rMax[r], acc[r]);
    }
    #pragma unroll
    for (int r = 0; r < 8; ++r) rMax[r] = half16_max(rMax[r]);  // true row max, all lanes

    // ---- pass 2: fold -rowMax into C init; per tile: wmma + 8 exp + 8 add ----
    v8f cfv;
    #pragma unroll
    for (int r = 0; r < 8; ++r) cfv[r] = afv[r] - rMax[r];

    float rSum[8];
    #pragma unroll
    for (int r = 0; r < 8; ++r) rSum[r] = 0.0f;

    for (int t = 0; t < TILES; ++t) {
        const v2f Bm = bview[bidx0 + 32 * t];
        const v8f acc = __builtin_amdgcn_wmma_f32_16x16x4_f32(false, A, false, Bm,
                                                              (short)0, cfv, false, false);
        #pragma unroll
        for (int r = 0; r < 8; ++r) rSum[r] += __expf(acc[r]);  // acc = m - rowMax <= 0
    }
    #pragma unroll
    for (int r = 0; r < 8; ++r) rSum[r] = half16_sum(rSum[r]);

    #pragma unroll
    for (int r = 0; r < 8; ++r) {
        const float lse = rMax[r] + __logf(rSum[r]);
        const float upd = EPSV * (LOG_MU - lse);
        if (lane == 0)  u[i0 + r]     += upd;
        if (lane == 16) u[i0 + 8 + r] += upd;
    }
}

// cost partials: sum_ij exp((u_i + v_j - C_ij)/eps) * C_ij.
// WMMA: A = (-2x0,-2x1 | -2x2, 1), B = (y0,y1 | y2, |y|^2), C init = |x|^2  =>  D = C_ij.
__global__ __launch_bounds__(THREADS_PB)
void emd_cost(const float* __restrict__ X, const float* __restrict__ Y,
              const float* __restrict__ U, const float* __restrict__ V,
              float* __restrict__ partials) {
    __shared__ float4 sB[NN];   // (y0, y1 | y2, |y|^2)
    __shared__ float  sv[NN];
    __shared__ float  sAcc[WAVES_PB];
    const int b = blockIdx.y;
    const float* R  = X + (size_t)b * DD * NN;
    const float* Cp = Y + (size_t)b * DD * NN;
    const float* u  = U + (size_t)b * NN;
    const float* v  = V + (size_t)b * NN;

    for (int j = threadIdx.x; j < NN; j += blockDim.x) {
        float y0 = Cp[j], y1 = Cp[NN + j], y2 = Cp[2 * NN + j];
        sB[j] = make_float4(y0, y1, y2, y0 * y0 + y1 * y1 + y2 * y2);
        sv[j] = v[j];
    }
    __syncthreads();

    const int lane = threadIdx.x & 31;
    const int wave = threadIdx.x >> 5;
    const int i0   = blockIdx.x * ROWS_PB + wave * 16;
    const bool hi  = lane >= 16;
    const int rl   = lane & 15;

    const v2f* bview = (const v2f*)sB;
    const int  bidx0 = 2 * rl + (hi ? 1 : 0);

    v2f A;
    v8f xnv;                       // C init = |x_i|^2
    float ufie[8];                 // u_i / eps
    {
        const int i = i0 + rl;
        const float x0 = R[i], x1 = R[NN + i], x2 = R[2 * NN + i];
        if (!hi) { A.x = -2.0f * x0; A.y = -2.0f * x1; }
        else     { A.x = -2.0f * x2; A.y = 1.0f; }
        const float xn   = x0 * x0 + x1 * x1 + x2 * x2;
        const float uval = u[i] * INV_EPS;
        #pragma unroll
        for (int r = 0; r < 8; ++r) {
            ufie[r] = __shfl(uval, hi ? (r + 8) : r, 32);
            xnv[r]  = __shfl(xn,   hi ? (r + 8) : r, 32);
        }
    }

    float accum = 0.0f;
    for (int t = 0; t < TILES; ++t) {
        const v2f Bm = bview[bidx0 + 32 * t];
        const float vvie = sv[t * 16 + rl] * INV_EPS;
        const v8f acc = __builtin_amdgcn_wmma_f32_16x16x4_f32(false, A, false, Bm,
                                                              (short)0, xnv, false, false);
        #pragma unroll
        for (int r = 0; r < 8; ++r) {
            const float Cij = acc[r];
            const float m   = ufie[r] + vvie - Cij * INV_EPS;
            accum += __expf(m) * Cij;
        }
    }
    accum += __shfl_xor(accum, 1, 32);
    accum += __shfl_xor(accum, 2, 32);
    accum += __shfl_xor(accum, 4, 32);
    accum += __shfl_xor(accum, 8, 32);
    accum += __shfl_xor(accum, 16, 32);
    if (lane == 0) sAcc[wave] = accum;
    __syncthreads();
    if (threadIdx.x == 0) {
        float s = 0.0f;
        #pragma unroll
        for (int w = 0; w < WAVES_PB; ++w) s += sAcc[w];
        partials[blockIdx.y * gridDim.x + blockIdx.x] = s;
    }
}

__global__ void emd_reduce(const float* __restrict__ partials, float* __restrict__ out, int n) {
    __shared__ float s[512];
    s[threadIdx.x] = (threadIdx.x < n) ? partials[threadIdx.x] : 0.0f;
    __syncthreads();
    if (threadIdx.x == 0) {
        float t = 0.0f;
        for (int i = 0; i < n && i < 512; ++i) t += s[i];
        out[0] = t / (float)BB;    // mean over batches
    }
}

extern "C" void kernel_launch(void* const* d_in, const int* in_sizes, int n_in,
                              void* d_out, int out_size, void* d_ws, size_t ws_size,
                              hipStream_t stream) {
    (void)in_sizes; (void)n_in; (void)out_size; (void)ws_size;
    const float* x = (const float*)d_in[0];     // [B, D, N]
    const float* y = (const float*)d_in[1];     // [B, D, N]
    float* ws       = (float*)d_ws;
    float* u        = ws;                       // [B, N]
    float* v        = ws + (size_t)BB * NN;     // [B, N]
    float* partials = ws + (size_t)2 * BB * NN; // [gridX * B]

    const int uvN = 2 * BB * NN;
    emd_init<<<(uvN + 255) / 256, 256, 0, stream>>>(ws, uvN);

    dim3 grid(NN / ROWS_PB, BB);                // (32, 8)
    for (int it = 0; it < MAX_ITER; ++it) {
        sinkhorn_half_pass<<<grid, THREADS_PB, 0, stream>>>(x, y, u, v); // u-update
        sinkhorn_half_pass<<<grid, THREADS_PB, 0, stream>>>(y, x, v, u); // v-update
    }
    emd_cost<<<grid, THREADS_PB, 0, stream>>>(x, y, u, v, partials);
    emd_reduce<<<1, 512, 0, stream>>>(partials, (float*)d_out, grid.x * BB);
}